// AdaptiveGroupNorm_73615739453611
// MI455X (gfx1250) — compile-verified
//
#include <hip/hip_runtime.h>
#include <hip/hip_bf16.h>
#include <math.h>

typedef __attribute__((ext_vector_type(2))) float v2f;
typedef __attribute__((ext_vector_type(4))) float v4f;
typedef __attribute__((ext_vector_type(8))) float v8f;

// d_ws layout:
//   bytes [0, 16)   : double sum, double sumsq
//   bytes [64, ...) : row coverage bitmap (1 bit per row of xf)

__global__ void agn_init_ws(unsigned* __restrict__ ws_words, int n_words) {
    for (int i = threadIdx.x; i < n_words; i += blockDim.x) ws_words[i] = 0u;
}

__global__ void agn_mark(const int* __restrict__ indexes, int n_idx,
                         unsigned* __restrict__ bitmap) {
    int i = blockIdx.x * blockDim.x + threadIdx.x;
    if (i < n_idx) {
        unsigned r = (unsigned)indexes[i];
        atomicOr(&bitmap[r >> 5], 1u << (r & 31u));
    }
}

// Global sum / sum-of-squares via V_WMMA_F32_16X16X4_F32 with B = ones.
// Each wave consumes 128 contiguous floats per step: one global_load_b128 per
// lane (v4f), split into two 16x4 A-tiles. With B = ones, D[m][n] = sum_k A[m][k],
// so the A-slot<->memory mapping is irrelevant for a total sum. Every D column
// replicates the row sums, so the sum over all 256 D elements is 16x the true
// sum; we correct with an exact *1/16 at the end.
__global__ void agn_stats(const float* __restrict__ x, unsigned long long n,
                          double* __restrict__ sums) {
    const int lane = threadIdx.x & 31;
    const int wave = threadIdx.x >> 5;
    const int wpb  = blockDim.x >> 5;
    const unsigned long long waveId = (unsigned long long)blockIdx.x * wpb + wave;
    const unsigned long long nWaves = (unsigned long long)gridDim.x * wpb;

    v8f accS = {0.f,0.f,0.f,0.f,0.f,0.f,0.f,0.f};
    v8f accQ = {0.f,0.f,0.f,0.f,0.f,0.f,0.f,0.f};
    v2f ones; ones[0] = 1.0f; ones[1] = 1.0f;

    const unsigned long long nfull = n & ~127ull;
    for (unsigned long long base = waveId * 128ull; base + 128ull <= n;
         base += nWaves * 128ull) {
        v4f a = *(const v4f*)(x + base + (unsigned)(lane << 2));   // b128
        v4f a2 = a * a;                                            // 2x v_pk_mul_f32
        v2f lo, hi, lo2, hi2;
        lo[0]  = a[0];  lo[1]  = a[1];  hi[0]  = a[2];  hi[1]  = a[3];
        lo2[0] = a2[0]; lo2[1] = a2[1]; hi2[0] = a2[2]; hi2[1] = a2[3];
        accS = __builtin_amdgcn_wmma_f32_16x16x4_f32(
            false, lo,  false, ones, (short)0, accS, false, false);
        accS = __builtin_amdgcn_wmma_f32_16x16x4_f32(
            false, hi,  false, ones, (short)0, accS, false, false);
        accQ = __builtin_amdgcn_wmma_f32_16x16x4_f32(
            false, lo2, false, ones, (short)0, accQ, false, false);
        accQ = __builtin_amdgcn_wmma_f32_16x16x4_f32(
            false, hi2, false, ones, (short)0, accQ, false, false);
    }

    float s = 0.f, q = 0.f;
#pragma unroll
    for (int i = 0; i < 8; ++i) { s += accS[i]; q += accQ[i]; }

    // Tail (n % 128 != 0): scalar, scaled by 16 to match WMMA replication.
    if (waveId == 0) {
        for (unsigned long long i = nfull + (unsigned)lane; i < n; i += 32ull) {
            float v = x[i];
            s += v * 16.0f;
            q += v * v * 16.0f;
        }
    }

#pragma unroll
    for (int m = 16; m >= 1; m >>= 1) {
        s += __shfl_xor(s, m, 32);
        q += __shfl_xor(q, m, 32);
    }

    __shared__ double lsum[16];  // up to 8 waves * {s,q}
    if (lane == 0) { lsum[wave * 2] = (double)s; lsum[wave * 2 + 1] = (double)q; }
    __syncthreads();
    if (threadIdx.x == 0) {
        double ts = 0.0, tq = 0.0;
        for (int i = 0; i < wpb; ++i) { ts += lsum[i * 2]; tq += lsum[i * 2 + 1]; }
        atomicAdd(&sums[0], ts * 0.0625);   // undo 16x column replication
        atomicAdd(&sums[1], tq * 0.0625);
    }
}

// One block per row. Covered rows: out = x*(inv*w) + (b - mean*inv*w).
// Uncovered rows: out = b. Non-temporal stores keep x resident in L2 for the
// second pass (the whole 128 MiB tensor fits in the 192 MB L2).
__global__ void agn_norm(const float* __restrict__ x,
                         const float* __restrict__ weight,
                         const float* __restrict__ bias,
                         const unsigned* __restrict__ bitmap,
                         const double* __restrict__ sums,
                         float* __restrict__ out,
                         int C, int HW, long long cnt, float eps) {
    const int r = blockIdx.x;
    const double mean = sums[0] / (double)cnt;
    const double var  = (sums[1] - (double)cnt * mean * mean) / (double)(cnt - 1);
    const float  inv  = (float)(1.0 / sqrt(var + (double)eps));
    const float  fm   = (float)mean;

    const int   c  = r % C;
    const float wc = weight[c] * inv;
    const float sh = bias[c] - fm * wc;
    const float bc = bias[c];
    const bool covered = (bitmap[r >> 5] >> (r & 31)) & 1u;

    const v4f* xp = (const v4f*)(x + (size_t)r * (size_t)HW);
    v4f* op = (v4f*)(out + (size_t)r * (size_t)HW);
    const int nvec = HW >> 2;

    if (covered) {
        for (int j = threadIdx.x; j < nvec; j += blockDim.x) {
            v4f v = xp[j];
            v4f o;
            o[0] = v[0] * wc + sh;
            o[1] = v[1] * wc + sh;
            o[2] = v[2] * wc + sh;
            o[3] = v[3] * wc + sh;
            __builtin_nontemporal_store(o, op + j);
        }
    } else {
        v4f o; o[0] = bc; o[1] = bc; o[2] = bc; o[3] = bc;
        for (int j = threadIdx.x; j < nvec; j += blockDim.x)
            __builtin_nontemporal_store(o, op + j);
    }
}

extern "C" void kernel_launch(void* const* d_in, const int* in_sizes, int n_in,
                              void* d_out, int out_size, void* d_ws, size_t ws_size,
                              hipStream_t stream) {
    const float* x   = (const float*)d_in[0];
    const float* w   = (const float*)d_in[1];
    const float* b   = (const float*)d_in[2];
    const int*   idx = (const int*)d_in[3];
    float* out = (float*)d_out;

    const long long n = (long long)in_sizes[0];   // total elements (N*C*H*W)
    const int C     = in_sizes[1];                // weight has C elements
    const int n_idx = in_sizes[3];                // N*C rows
    const int rows  = n_idx;
    const int HW    = (int)(n / rows);

    double*   sums   = (double*)d_ws;
    unsigned* bitmap = (unsigned*)((char*)d_ws + 64);
    const int bitmapWords = (rows + 31) / 32;
    const int n_words = 16 + bitmapWords;         // sums (16 words incl. pad) + bitmap

    agn_init_ws<<<1, 1024, 0, stream>>>((unsigned*)d_ws, n_words);
    agn_mark<<<(n_idx + 255) / 256, 256, 0, stream>>>(idx, n_idx, bitmap);
    agn_stats<<<2048, 256, 0, stream>>>(x, (unsigned long long)n, sums);
    agn_norm<<<rows, 256, 0, stream>>>(x, w, b, bitmap, sums, out, C, HW, n, 1e-5f);
}